// BatchRemoveQuatDiscontinuities_38482906972730
// MI455X (gfx1250) — compile-verified
//
#include <hip/hip_runtime.h>

// Problem constants (B clips, T frames, J joints, quat=4 floats)
#define Bq 128
#define Tq 1024
#define Jq 64
#define NC 16              // chunks along T
#define TC (Tq / NC)       // 64 time steps per chunk
#define TT 8               // time rows staged per async tile
#define NTILE (TC / TT)    // 8 tiles per chunk

typedef float v4f __attribute__((ext_vector_type(4)));

// ---- CDNA5 async global->LDS copy (VGLOBAL encoding, tracked by ASYNCcnt) ----
__device__ __forceinline__ void async_copy_row(unsigned lds_addr, const float4* gaddr) {
  asm volatile("global_load_async_to_lds_b128 %0, %1, off"
               :: "v"(lds_addr), "v"((unsigned long long)gaddr)
               : "memory");
}

template <int N>
__device__ __forceinline__ void wait_asynccnt() {
  asm volatile("s_wait_asynccnt %0" :: "n"(N) : "memory");
}

__device__ __forceinline__ unsigned lds_off(const void* p) {
  // flat shared pointer low 32 bits == wave-relative LDS byte offset
  return (unsigned)(unsigned long long)p;
}

// ---------------------------------------------------------------------------
// Pass 1: per (b, chunk, j) compute flip parity of the chunk.
// grid = (NC, Bq), block = 64 (lane = j) -> 1KB contiguous per time row.
// Default RT cache policy: leaves the 128 MiB input resident in the 192 MB L2
// so pass 2's re-read mostly hits L2.
// ---------------------------------------------------------------------------
__global__ void __launch_bounds__(64)
parity_pass(const float* __restrict__ qf, unsigned char* __restrict__ par) {
  __shared__ float4 tile[2][TT][Jq];     // 2 x 8KB double buffer
  const int j = threadIdx.x;
  const int k = blockIdx.x;              // chunk
  const int b = blockIdx.y;
  const float4* qb = reinterpret_cast<const float4*>(qf) + (size_t)b * Tq * Jq;
  const int t0 = k * TC;

  // prologue: issue tile 0
#pragma unroll
  for (int r = 0; r < TT; ++r)
    async_copy_row(lds_off(&tile[0][r][j]), qb + (size_t)(t0 + r) * Jq + j);

  // previous quaternion (q[t0-1], or q[0] for chunk 0 which skips t=0)
  float4 prev = qb[(size_t)(t0 == 0 ? 0 : t0 - 1) * Jq + j];
  unsigned parity = 0u;

  for (int ti = 0; ti < NTILE; ++ti) {
    const int cb = ti & 1;
    if (ti + 1 < NTILE) {
#pragma unroll
      for (int r = 0; r < TT; ++r)
        async_copy_row(lds_off(&tile[cb ^ 1][r][j]),
                       qb + (size_t)(t0 + (ti + 1) * TT + r) * Jq + j);
      wait_asynccnt<TT>();   // next tile may remain in flight; current tile done
    } else {
      wait_asynccnt<0>();
    }
    const int tb = t0 + ti * TT;
#pragma unroll
    for (int r = 0; r < TT; ++r) {
      float4 cur = tile[cb][r][j];       // ds_load_b128
      const int t = tb + r;
      if (t != 0) {
        float d = prev.x * cur.x + prev.y * cur.y + prev.z * cur.z + prev.w * cur.w;
        parity ^= (d < 0.0f) ? 1u : 0u;
      }
      prev = cur;
    }
  }
  par[((size_t)b * NC + k) * Jq + j] = (unsigned char)parity;
}

// ---------------------------------------------------------------------------
// Pass 2: re-stream q (L2-resident), seed parity with the XOR of all preceding
// chunk parities, recompute local dots, apply sign by XOR-ing the float sign
// bit (exact for multiplication by +-1). Output stores are non-temporal so the
// 128 MiB of writes do not evict the input from L2.
// ---------------------------------------------------------------------------
__global__ void __launch_bounds__(64)
apply_pass(const float* __restrict__ qf, const unsigned char* __restrict__ par,
           float* __restrict__ outf) {
  __shared__ float4 tile[2][TT][Jq];
  const int j = threadIdx.x;
  const int k = blockIdx.x;
  const int b = blockIdx.y;
  const float4* qb = reinterpret_cast<const float4*>(qf) + (size_t)b * Tq * Jq;
  float* ofb = outf + (size_t)b * Tq * Jq * 4;
  const int t0 = k * TC;

#pragma unroll
  for (int r = 0; r < TT; ++r)
    async_copy_row(lds_off(&tile[0][r][j]), qb + (size_t)(t0 + r) * Jq + j);

  float4 prev = qb[(size_t)(t0 == 0 ? 0 : t0 - 1) * Jq + j];

  // exclusive prefix-XOR of chunk parities (<= 15 byte loads, L2/WGP$ hot)
  unsigned parity = 0u;
  for (int kk = 0; kk < k; ++kk)
    parity ^= (unsigned)par[((size_t)b * NC + kk) * Jq + j];

  for (int ti = 0; ti < NTILE; ++ti) {
    const int cb = ti & 1;
    if (ti + 1 < NTILE) {
#pragma unroll
      for (int r = 0; r < TT; ++r)
        async_copy_row(lds_off(&tile[cb ^ 1][r][j]),
                       qb + (size_t)(t0 + (ti + 1) * TT + r) * Jq + j);
      wait_asynccnt<TT>();
    } else {
      wait_asynccnt<0>();
    }
    const int tb = t0 + ti * TT;
#pragma unroll
    for (int r = 0; r < TT; ++r) {
      float4 cur = tile[cb][r][j];
      const int t = tb + r;
      if (t != 0) {
        float d = prev.x * cur.x + prev.y * cur.y + prev.z * cur.z + prev.w * cur.w;
        parity ^= (d < 0.0f) ? 1u : 0u;
      }
      prev = cur;
      const unsigned m = parity << 31;   // sign-bit mask: flips sign iff parity==1
      v4f o;
      o.x = __uint_as_float(__float_as_uint(cur.x) ^ m);
      o.y = __uint_as_float(__float_as_uint(cur.y) ^ m);
      o.z = __uint_as_float(__float_as_uint(cur.z) ^ m);
      o.w = __uint_as_float(__float_as_uint(cur.w) ^ m);
      // non-temporal 128-bit store: don't pollute L2 (keep input resident)
      __builtin_nontemporal_store(
          o, reinterpret_cast<v4f*>(ofb + ((size_t)t * Jq + j) * 4));
    }
  }
}

extern "C" void kernel_launch(void* const* d_in, const int* in_sizes, int n_in,
                              void* d_out, int out_size, void* d_ws, size_t ws_size,
                              hipStream_t stream) {
  (void)in_sizes; (void)n_in; (void)out_size; (void)ws_size;
  const float* q = (const float*)d_in[0];
  float* out = (float*)d_out;
  unsigned char* par = (unsigned char*)d_ws;   // NC*Bq*Jq = 128 KiB scratch

  dim3 grid(NC, Bq), blk(64);
  parity_pass<<<grid, blk, 0, stream>>>(q, par);
  apply_pass<<<grid, blk, 0, stream>>>(q, par, out);
}